// SGCluster_29137058136183
// MI455X (gfx1250) — compile-verified
//
#include <hip/hip_runtime.h>

#define DIM 128
#define LN_EPS 1e-5f

typedef float v2f __attribute__((ext_vector_type(2)));
typedef float v8f __attribute__((ext_vector_type(8)));

// ---------------------------------------------------------------------------
// 1) LayerNorm: one wave32 per row, 4 floats per lane, shfl_xor reductions.
// ---------------------------------------------------------------------------
__global__ __launch_bounds__(256) void ln_kernel(const float* __restrict__ x,
                                                 const float* __restrict__ w,
                                                 const float* __restrict__ b,
                                                 float* __restrict__ y, int n) {
    const int wave = threadIdx.x >> 5;
    const int lane = threadIdx.x & 31;
    const int row  = blockIdx.x * 8 + wave;
    if (row >= n) return;

    float4 v = ((const float4*)(x + (size_t)row * DIM))[lane];
    float s = v.x + v.y + v.z + v.w;
    #pragma unroll
    for (int off = 16; off > 0; off >>= 1) s += __shfl_xor(s, off, 32);
    const float mu = s * (1.0f / DIM);

    float4 d = make_float4(v.x - mu, v.y - mu, v.z - mu, v.w - mu);
    float q = d.x * d.x + d.y * d.y + d.z * d.z + d.w * d.w;
    #pragma unroll
    for (int off = 16; off > 0; off >>= 1) q += __shfl_xor(q, off, 32);
    const float inv = rsqrtf(q * (1.0f / DIM) + LN_EPS);

    const float4 wv = ((const float4*)w)[lane];
    const float4 bv = ((const float4*)b)[lane];
    float4 o;
    o.x = d.x * inv * wv.x + bv.x;
    o.y = d.y * inv * wv.y + bv.y;
    o.z = d.z * inv * wv.z + bv.z;
    o.w = d.w * inv * wv.w + bv.w;
    ((float4*)(y + (size_t)row * DIM))[lane] = o;
}

// ---------------------------------------------------------------------------
// 2) Degree: init to 1.0 (self-loop), atomic +1 per incoming edge, then rsqrt.
// ---------------------------------------------------------------------------
__global__ __launch_bounds__(256) void fill1_kernel(float* __restrict__ deg, int n) {
    int i = blockIdx.x * blockDim.x + threadIdx.x;
    if (i < n) deg[i] = 1.0f;
}

__global__ __launch_bounds__(256) void deg_kernel(const int* __restrict__ dst,
                                                  float* __restrict__ deg, int e) {
    int i = blockIdx.x * blockDim.x + threadIdx.x;
    if (i < e) atomicAdd(&deg[dst[i]], 1.0f);
}

__global__ __launch_bounds__(256) void rsqrt_kernel(float* __restrict__ deg, int n) {
    int i = blockIdx.x * blockDim.x + threadIdx.x;
    if (i < n) deg[i] = rsqrtf(deg[i]);   // deg >= 1 always (self-loop)
}

// ---------------------------------------------------------------------------
// 3) Propagation. Self-loop term initializes the output buffer, then edges
//    scatter-add. 32 lanes cover one 512B row (float4 per lane).
// ---------------------------------------------------------------------------
__global__ __launch_bounds__(256) void prop_self_kernel(const float* __restrict__ dinv,
                                                        const float* __restrict__ xin,
                                                        float* __restrict__ xout, int n) {
    long long t = (long long)blockIdx.x * blockDim.x + threadIdx.x;
    int i = (int)(t >> 5);
    if (i >= n) return;
    int q = (int)(t & 31);
    float di = dinv[i];
    float wl = di * di;
    float4 v = *(const float4*)(xin + (size_t)i * DIM + q * 4);
    float4 o = make_float4(wl * v.x, wl * v.y, wl * v.z, wl * v.w);
    *(float4*)(xout + (size_t)i * DIM + q * 4) = o;
}

__global__ __launch_bounds__(256) void prop_edge_kernel(const int* __restrict__ src,
                                                        const int* __restrict__ dst,
                                                        const float* __restrict__ dinv,
                                                        const float* __restrict__ xin,
                                                        float* __restrict__ xout, int e) {
    long long t = (long long)blockIdx.x * blockDim.x + threadIdx.x;
    int ei = (int)(t >> 5);
    if (ei >= e) return;
    int q = (int)(t & 31);
    int s = src[ei];
    int d = dst[ei];
    float w = dinv[s] * dinv[d];
    float4 v = *(const float4*)(xin + (size_t)s * DIM + q * 4);
    float* o = xout + (size_t)d * DIM + q * 4;
    atomicAdd(o + 0, w * v.x);
    atomicAdd(o + 1, w * v.y);
    atomicAdd(o + 2, w * v.z);
    atomicAdd(o + 3, w * v.w);
}

// ---------------------------------------------------------------------------
// 4) GEMM: out[m,n] = sum_k y[m,k] * W[n,k] + b[n], via V_WMMA_F32_16X16X4_F32.
//    Block = 256 threads = 8 waves; block covers 16 rows x 128 cols; each wave
//    owns one 16x16 tile (col tile = wave id). K-loop: 32 WMMA steps of K=4.
//    A-lane layout: lane (half=L>=16, l=L&15) -> A[l][k0+2*half+{0,1}].
//    B-lane layout: lane -> B[k0+2*half+{0,1}][l] = W[l][k0+2*half+{0,1}].
//    D layout: acc[r] -> M = r + 8*half, N = l.
// ---------------------------------------------------------------------------
__global__ __launch_bounds__(256) void gemm_wmma_kernel(const float* __restrict__ y,
                                                        const float* __restrict__ W,
                                                        const float* __restrict__ bias,
                                                        float* __restrict__ out,
                                                        int ntile /* rows/16 */) {
    const int waveId = threadIdx.x >> 5;
    const int lane   = threadIdx.x & 31;
    const int half   = lane >> 4;
    const int l      = lane & 15;
    const int row0   = blockIdx.x * 16;
    const int col0   = waveId * 16;
    (void)ntile;

    const float* arow = y + (size_t)(row0 + l) * DIM + 2 * half;  // y[m][k]
    const float* bcol = W + (size_t)(col0 + l) * DIM + 2 * half;  // W[n][k]

    v8f acc = {};
    #pragma unroll
    for (int kk = 0; kk < 32; ++kk) {
        v2f a = *(const v2f*)(arow + 4 * kk);
        v2f b = *(const v2f*)(bcol + 4 * kk);
        acc = __builtin_amdgcn_wmma_f32_16x16x4_f32(
            /*neg_a=*/false, a, /*neg_b=*/false, b,
            /*c_mod=*/(short)0, acc, /*reuse_a=*/false, /*reuse_b=*/false);
    }

    const int   ncol = col0 + l;
    const float bv   = bias[ncol];
    #pragma unroll
    for (int r = 0; r < 8; ++r) {
        int mrow = row0 + r + 8 * half;
        out[(size_t)mrow * DIM + ncol] = acc[r] + bv;
    }
}

// Scalar fallback for a row-remainder (< 16 rows). Not used when N % 16 == 0.
__global__ __launch_bounds__(128) void gemm_tail_kernel(const float* __restrict__ y,
                                                        const float* __restrict__ W,
                                                        const float* __restrict__ bias,
                                                        float* __restrict__ out,
                                                        int row_start, int n) {
    int row = row_start + blockIdx.x;
    if (row >= n) return;
    int col = threadIdx.x;
    const float* yr = y + (size_t)row * DIM;
    const float* wr = W + (size_t)col * DIM;
    float acc = 0.0f;
    #pragma unroll 8
    for (int k = 0; k < DIM; ++k) acc += yr[k] * wr[k];
    out[(size_t)row * DIM + col] = acc + bias[col];
}

// ---------------------------------------------------------------------------
extern "C" void kernel_launch(void* const* d_in, const int* in_sizes, int n_in,
                              void* d_out, int out_size, void* d_ws, size_t ws_size,
                              hipStream_t stream) {
    const float* x    = (const float*)d_in[0];
    const int*   ei   = (const int*)d_in[1];   // edge_index [2, E] (int32 under default JAX x32)
    const float* ln_w = (const float*)d_in[2];
    const float* ln_b = (const float*)d_in[3];
    const float* W    = (const float*)d_in[4];
    const float* bias = (const float*)d_in[5];

    const int N = in_sizes[0] / DIM;
    const int E = in_sizes[1] / 2;
    const int* src = ei;
    const int* dst = ei + E;

    float* bufA = (float*)d_ws;                    // LN output, then hop-2 output
    float* bufB = bufA + (size_t)N * DIM;          // hop-1 output
    float* dinv = bufB + (size_t)N * DIM;          // deg -> dinv, N floats

    float* out = (float*)d_out;

    // 1) LayerNorm: x -> bufA
    ln_kernel<<<(N + 7) / 8, 256, 0, stream>>>(x, ln_w, ln_b, bufA, N);

    // 2) degree + dinv (in place in `dinv`)
    fill1_kernel<<<(N + 255) / 256, 256, 0, stream>>>(dinv, N);
    deg_kernel<<<(E + 255) / 256, 256, 0, stream>>>(dst, dinv, E);
    rsqrt_kernel<<<(N + 255) / 256, 256, 0, stream>>>(dinv, N);

    const int nodeBlocks = (int)(((long long)N * 32 + 255) / 256);
    const int edgeBlocks = (int)(((long long)E * 32 + 255) / 256);

    // 3) hop 1: bufA -> bufB
    prop_self_kernel<<<nodeBlocks, 256, 0, stream>>>(dinv, bufA, bufB, N);
    prop_edge_kernel<<<edgeBlocks, 256, 0, stream>>>(src, dst, dinv, bufA, bufB, E);

    //    hop 2: bufB -> bufA
    prop_self_kernel<<<nodeBlocks, 256, 0, stream>>>(dinv, bufB, bufA, N);
    prop_edge_kernel<<<edgeBlocks, 256, 0, stream>>>(src, dst, dinv, bufB, bufA, E);

    // 4) Linear via f32 WMMA: bufA @ W^T + b -> out
    const int ntile = N / 16;
    if (ntile > 0)
        gemm_wmma_kernel<<<ntile, 256, 0, stream>>>(bufA, W, bias, out, ntile);
    const int rem = N - ntile * 16;
    if (rem > 0)
        gemm_tail_kernel<<<rem, 128, 0, stream>>>(bufA, W, bias, out, ntile * 16, N);
}